// GeometricAttention_88029649699077
// MI455X (gfx1250) — compile-verified
//
#include <hip/hip_runtime.h>
#include <math.h>

// ---------------------------------------------------------------------------
// GeometricAttention for MI455X (gfx1250)
//
// Roofline: attn output = 8192^2 * 4B = 256MB HBM write (~11us @ 23.3TB/s) is
// the floor; theta/tau (32KB each) are L2/LDS-resident. Pass 1 fuses score
// generation with row-sum + the three attn@v reductions using
// V_WMMA_F32_16X16X4_F32 (C columns 0..3 accumulate Rx, Ry, T, S per row).
// Pass 2 recomputes scores (cheap vs the write) and streams normalized attn
// with non-temporal b128 stores. Inner loops are branchless (EXEC all-1s for
// WMMA) and use native v_exp_f32 to keep VALU under the HBM floor.
// ---------------------------------------------------------------------------

#define N_PTS    8192
#define ROWS     16      // rows per block (one WMMA M-tile)
#define CHUNK    2048    // columns staged in LDS at a time
#define NTHREADS 256     // 8 waves (wave32)
#define NWAVES   8

typedef float v2f __attribute__((ext_vector_type(2)));
typedef float v4f __attribute__((ext_vector_type(4)));
typedef float v8f __attribute__((ext_vector_type(8)));

__device__ __forceinline__ float geo_score(float ti, float taui,
                                           float tj, float tauj, bool diag) {
    const float TWO_PI = 6.28318530717958647692f;
    float diff = fabsf(ti - tj);                      // abs folds to src mod
    float dth  = fminf(diff, TWO_PI - diff);
    float dist = fmaf(0.3f, fabsf(taui - tauj), dth); // TAU_W = 0.3
    float s    = __expf(-2.0f * dist);                // LAMBDA = 2, native exp
    return diag ? 0.0f : s;                           // one v_cndmask
}

__global__ __launch_bounds__(NTHREADS)
void geo_attn_kernel(const float* __restrict__ theta,
                     const float* __restrict__ tau,
                     float* __restrict__ theta_out,
                     float* __restrict__ tau_out,
                     float* __restrict__ attn) {
    __shared__ v2f   sTT[CHUNK];          // (theta, tau) interleaved: 16KB
    __shared__ float sB[4 * CHUNK];       // rows: cos | sin | tau | ones: 32KB
    __shared__ float sAcc[NWAVES][8][32]; // per-wave WMMA C dumps
    __shared__ float sRow[ROWS][4];       // Rx, Ry, T, S per row
    __shared__ float sInvS[ROWS];

    const int tid     = threadIdx.x;
    const int wave    = tid >> 5;
    const int lane    = tid & 31;
    const int rowBase = blockIdx.x * ROWS;

    // WMMA 16x16x4 f32 A/B lane mapping (ISA 7.12.2):
    //   lanes 0-15:  v0 -> K=0, v1 -> K=1 ; lanes 16-31: v0 -> K=2, v1 -> K=3
    //   A row index M = lane&15 ; B col index N = lane&15
    const int m  = lane & 15;
    const int k0 = (lane >> 4) * 2;

    // Per-lane B source: row of sB (cos/sin/tau/ones), zeroed for n > 3.
    const int   laneOff = ((m < 3) ? m : 3) * CHUNK;
    const float bscale  = (m < 4) ? 1.0f : 0.0f;

    const int   row  = rowBase + m;
    const float ti   = theta[row];
    const float taui = tau[row];

    v8f c = {0.f, 0.f, 0.f, 0.f, 0.f, 0.f, 0.f, 0.f};

    for (int c0 = 0; c0 < N_PTS; c0 += CHUNK) {
        // Stage column data; cos/sin computed once per column (precise libm —
        // negligible count, feeds the validated Rx/Ry reductions).
        for (int idx = tid; idx < CHUNK; idx += NTHREADS) {
            float t = theta[c0 + idx];
            float u = tau[c0 + idx];
            v2f tt; tt.x = t; tt.y = u;
            sTT[idx]             = tt;
            sB[idx]              = cosf(t);
            sB[CHUNK + idx]      = sinf(t);
            sB[2 * CHUNK + idx]  = u;
            sB[3 * CHUNK + idx]  = 1.0f;
        }
        __syncthreads();

        // Each wave owns a 256-column slice: 64 branchless WMMA steps of K=4.
        const int base = wave * (CHUNK / NWAVES);
        #pragma unroll 4
        for (int it = 0; it < CHUNK / NWAVES; it += 4) {
            const int l0  = base + it;
            const int jl0 = l0 + k0;            // even -> 16B-aligned loads
            const int gj0 = c0 + jl0;

            // One ds_load_b128: (th0, ta0, th1, ta1) for this lane's 2 cols.
            const v4f tt = *reinterpret_cast<const v4f*>(&sTT[jl0]);
            // One ds_load_b64: this lane's two B values.
            const v2f bb = *reinterpret_cast<const v2f*>(&sB[laneOff + jl0]);

            v2f a, b;
            a.x = geo_score(ti, taui, tt.x, tt.y, row == gj0);
            a.y = geo_score(ti, taui, tt.z, tt.w, row == gj0 + 1);
            b.x = bb.x * bscale;
            b.y = bb.y * bscale;

            // D = A(16x4) * B(4x16) + C ; columns 0..3 accumulate Rx,Ry,T,S.
            c = __builtin_amdgcn_wmma_f32_16x16x4_f32(
                    false, a, false, b, (short)0, c, false, false);
        }
        __syncthreads();   // before restaging LDS
    }

    // Dump per-wave accumulators and reduce across waves.
    #pragma unroll
    for (int v = 0; v < 8; ++v) sAcc[wave][v][lane] = c[v];
    __syncthreads();

    if (tid < ROWS * 4) {
        const int mm = tid >> 2;        // row within tile
        const int nn = tid & 3;         // 0:Rx 1:Ry 2:T 3:S
        // C/D layout: VGPR v holds M=v (lanes 0-15) and M=v+8 (lanes 16-31).
        const int vv = mm & 7;
        const int ll = (mm < 8) ? nn : (nn + 16);
        float s = 0.f;
        #pragma unroll
        for (int w = 0; w < NWAVES; ++w) s += sAcc[w][vv][ll];
        sRow[mm][nn] = s;
    }
    __syncthreads();

    if (tid < ROWS) {
        const float Rx = sRow[tid][0];
        const float Ry = sRow[tid][1];
        const float T  = sRow[tid][2];
        const float S  = sRow[tid][3];
        const float inv = 1.0f / S;
        sInvS[tid] = inv;
        // atan2(Ry/S, Rx/S) == atan2(Ry, Rx): normalization cancels (S > 0).
        theta_out[rowBase + tid] = atan2f(Ry, Rx);
        tau_out[rowBase + tid]   = T * inv;
    }
    __syncthreads();

    // Pass 2: recompute scores and stream normalized attn. Column data loaded
    // once per 4-col group (v4f), all 16 rows served from registers; 16
    // coalesced non-temporal b128 stores per group (attn >> L2, never reused).
    float rth[ROWS], rta[ROWS], rin[ROWS];
    #pragma unroll
    for (int r = 0; r < ROWS; ++r) {
        rth[r] = theta[rowBase + r];
        rta[r] = tau[rowBase + r];
        rin[r] = sInvS[r];
    }

    const v4f* __restrict__ th4p = reinterpret_cast<const v4f*>(theta);
    const v4f* __restrict__ ta4p = reinterpret_cast<const v4f*>(tau);

    for (int q = tid; q < N_PTS / 4; q += NTHREADS) {
        const v4f th4 = th4p[q];
        const v4f ta4 = ta4p[q];
        const int col = q * 4;
        #pragma unroll
        for (int r = 0; r < ROWS; ++r) {
            const int i = rowBase + r;
            v4f o;
            o.x = geo_score(rth[r], rta[r], th4.x, ta4.x, i == col)     * rin[r];
            o.y = geo_score(rth[r], rta[r], th4.y, ta4.y, i == col + 1) * rin[r];
            o.z = geo_score(rth[r], rta[r], th4.z, ta4.z, i == col + 2) * rin[r];
            o.w = geo_score(rth[r], rta[r], th4.w, ta4.w, i == col + 3) * rin[r];
            __builtin_nontemporal_store(
                o, reinterpret_cast<v4f*>(attn + (size_t)i * N_PTS) + q);
        }
    }
}

extern "C" void kernel_launch(void* const* d_in, const int* in_sizes, int n_in,
                              void* d_out, int out_size, void* d_ws, size_t ws_size,
                              hipStream_t stream) {
    const float* theta = (const float*)d_in[0];
    const float* tau   = (const float*)d_in[1];

    float* out       = (float*)d_out;
    float* theta_out = out;                 // [N]
    float* tau_out   = out + N_PTS;         // [N]
    float* attn      = out + 2 * N_PTS;     // [N, N]

    dim3 grid(N_PTS / ROWS);                // 512 blocks of 16 rows
    dim3 block(NTHREADS);
    geo_attn_kernel<<<grid, block, 0, stream>>>(theta, tau,
                                                theta_out, tau_out, attn);
}